// IGSO3_51402168599283
// MI455X (gfx1250) — compile-verified
//
#include <hip/hip_runtime.h>
#include <math.h>

// IGSO3 grad-log-pdf, MI455X (gfx1250, wave32).
// Phase 1: V_WMMA_F32_16X16X4_F32 builds series = expon @ sine_terms^T per
//          16-sigma row-group, branchless gauss-select + row-normalize in LDS,
//          emit pdf_pad[B,1002] into d_ws.
// Phase 2: per-row streaming gather/interp (pdf row cached in LDS), float4 IO.

#define B_ROWS   8192
#define N_OMEGA  4096
#define W_BINS   1000
#define L_TERMS  101
#define K_PAD    104            // 101 padded to multiple of 4
#define N_TILES  63             // ceil(1000/16)
#define PDF_STRIDE 1008
#define PAD_W    1002           // 1 + 1000 + 1

#define GAUSS_THRES      0.2f
#define GRAD_GAUSS_THRES 0.6f

typedef __attribute__((ext_vector_type(2))) float v2f;
typedef __attribute__((ext_vector_type(8))) float v8f;

__device__ __forceinline__ float bin_width_f() {
    return (float)(3.14159265358979323846 / 1000.0);
}

// ---------------- Kernel 1: pdf_pad builder (WMMA) -------------------------
// grid = B/16 blocks, 128 threads (4 waves). Each wave owns 16x16 tiles of
// the [16 x 1000] series matrix for this sigma group.
__global__ void __launch_bounds__(128)
igso3_pdf_kernel(const float* __restrict__ sigmas,
                 const float* __restrict__ omega_grid,
                 const float* __restrict__ sine_terms,   // [1000, 101] row-major
                 float* __restrict__ pdf_pad)            // [B, 1002]
{
    __shared__ float s_sig[16];
    __shared__ float s_expon[16 * K_PAD];        // A operand, zero-padded K
    __shared__ float s_pdf[16 * PDF_STRIDE];     // un-normalized pdf rows
    __shared__ float s_partial[16 * 8];
    __shared__ float s_inv[16];

    const int tid   = threadIdx.x;
    const int bbase = blockIdx.x * 16;           // first sigma row of group

    if (tid < 16) s_sig[tid] = sigmas[bbase + tid];
    __syncthreads();

    // expon[m,l] = exp(-0.5*l*(l+1)*sigma_m^2), zero for padded l
    for (int i = tid; i < 16 * K_PAD; i += 128) {
        const int m = i / K_PAD;
        const int l = i - m * K_PAD;
        float v = 0.0f;
        if (l < L_TERMS) {
            const float lf = (float)l;
            const float sg = s_sig[m];
            v = __expf(-0.5f * lf * (lf + 1.0f) * sg * sg);
        }
        s_expon[i] = v;
    }
    __syncthreads();

    const int lane  = tid & 31;
    const int wave  = tid >> 5;
    const int ncol  = lane & 15;     // N index inside tile (also A's M index)
    const int khalf = lane >> 4;     // selects K pair {0,1} vs {2,3}
    const int mbase = khalf * 8;

    // Loop-invariant per-lane row constants: sigma and -0.5/sigma^2
    float sgv[8], gc[8];
    #pragma unroll
    for (int r = 0; r < 8; ++r) {
        const float sg = s_sig[mbase + r];
        sgv[r] = sg;
        const float inv = 1.0f / sg;
        gc[r]  = -0.5f * inv * inv;
    }

    for (int t = wave; t < N_TILES; t += 4) {   // wave-uniform loop: EXEC all-1s
        v8f acc = {};
        const int wcol   = t * 16 + ncol;
        const int wclamp = (wcol < W_BINS) ? wcol : (W_BINS - 1);
        const float* __restrict__ brow = sine_terms + (size_t)wclamp * L_TERMS;

        #pragma unroll 2
        for (int kb = 0; kb < K_PAD; kb += 4) {
            const int l0 = kb + 2 * khalf;
            // A fragment (16x4): aligned float2 from LDS
            const v2f a = { s_expon[ncol * K_PAD + l0],
                            s_expon[ncol * K_PAD + l0 + 1] };
            // B fragment (4x16): two consecutive l of sine_terms row wcol
            const int lb0 = (l0     < L_TERMS) ? l0     : (L_TERMS - 1);
            const int lb1 = (l0 + 1 < L_TERMS) ? l0 + 1 : (L_TERMS - 1);
            const v2f b = { brow[lb0], brow[lb1] };
            acc = __builtin_amdgcn_wmma_f32_16x16x4_f32(
                false, a, false, b, (short)0, acc, false, false);
        }

        // Branchless gauss-select; only the store is predicated on wcol range.
        const float wmid = omega_grid[wclamp];
        const float w2   = wmid * wmid;
        if (wcol < W_BINS) {
            #pragma unroll
            for (int r = 0; r < 8; ++r) {
                const float gauss = __expf(w2 * gc[r]);
                const float val   = (sgv[r] > GAUSS_THRES) ? acc[r] : gauss;
                s_pdf[(mbase + r) * PDF_STRIDE + wcol] = val;
            }
        }
    }
    __syncthreads();

    // Row sums: 128 threads -> 16 rows x 8 chunks of 125 bins
    {
        const int row   = tid & 15;
        const int chunk = tid >> 4;
        float s = 0.0f;
        const float* p = &s_pdf[row * PDF_STRIDE + chunk * 125];
        #pragma unroll 5
        for (int w = 0; w < 125; ++w) s += p[w];
        s_partial[row * 8 + chunk] = s;
    }
    __syncthreads();
    if (tid < 16) {
        float tot = 0.0f;
        #pragma unroll
        for (int c = 0; c < 8; ++c) tot += s_partial[tid * 8 + c];
        s_inv[tid] = 1.0f / (tot * bin_width_f());
    }
    __syncthreads();

    // Emit normalized pdf_pad = [pdf[0], pdf[0..999], pdf[998]]
    for (int i = tid; i < 16 * PAD_W; i += 128) {
        const int m = i / PAD_W;
        const int j = i - m * PAD_W;
        const int src = (j == 0) ? 0 : ((j == PAD_W - 1) ? (W_BINS - 2) : (j - 1));
        pdf_pad[(size_t)(bbase + m) * PAD_W + j] =
            s_pdf[m * PDF_STRIDE + src] * s_inv[m];
    }
}

// ---------------- Kernel 2: gather + interpolate ---------------------------
// One block per sigma row; pdf_pad row cached in LDS; streams 4096 omegas as
// float4 (this kernel carries ~300 MB of the problem's HBM traffic).
__global__ void __launch_bounds__(256)
igso3_grad_kernel(const float* __restrict__ sigmas,
                  const float* __restrict__ omegas,     // [B, 4096]
                  const float* __restrict__ pdf_pad,    // [B, 1002]
                  float* __restrict__ out)              // [B, 4096]
{
    __shared__ float s_row[PAD_W];
    const int b   = blockIdx.x;
    const int tid = threadIdx.x;

    const float* __restrict__ prow = pdf_pad + (size_t)b * PAD_W;
    for (int i = tid; i < PAD_W; i += 256) s_row[i] = prow[i];

    const float sg      = sigmas[b];
    const bool  cond    = sg > GRAD_GAUSS_THRES;
    const float inv_s2  = 1.0f / (sg * sg);
    const float bw      = bin_width_f();
    __syncthreads();

    const float4* __restrict__ orow = (const float4*)(omegas + (size_t)b * N_OMEGA);
    float4*       __restrict__ drow = (float4*)(out + (size_t)b * N_OMEGA);

    for (int q = tid; q < N_OMEGA / 4; q += 256) {
        const float4 ov = orow[q];
        float om[4] = { ov.x, ov.y, ov.z, ov.w };
        float res[4];
        #pragma unroll
        for (int j = 0; j < 4; ++j) {
            int idx = (int)floorf(om[j] / bw + 0.5f);
            idx = idx < 0 ? 0 : (idx > W_BINS ? W_BINS : idx);
            const float shift = om[j] - ((float)idx + 0.5f) * bw;
            const float left  = s_row[idx];
            const float right = s_row[idx + 1];
            const float grad  = (right - left) / bw;
            const float den   = left + shift * grad;
            res[j] = cond ? (grad / den) : (-om[j] * inv_s2);
        }
        drow[q] = make_float4(res[0], res[1], res[2], res[3]);
    }
}

extern "C" void kernel_launch(void* const* d_in, const int* in_sizes, int n_in,
                              void* d_out, int out_size, void* d_ws, size_t ws_size,
                              hipStream_t stream) {
    const float* sigmas     = (const float*)d_in[0];   // [8192]
    const float* omegas     = (const float*)d_in[1];   // [8192, 4096]
    const float* omega_grid = (const float*)d_in[2];   // [1000]
    // d_in[3] = ls, unused on device
    const float* sine_terms = (const float*)d_in[4];   // [1000, 101]
    float* out     = (float*)d_out;
    float* pdf_pad = (float*)d_ws;                     // needs 8192*1002*4 = 32.8 MB

    hipLaunchKernelGGL(igso3_pdf_kernel, dim3(B_ROWS / 16), dim3(128), 0, stream,
                       sigmas, omega_grid, sine_terms, pdf_pad);
    hipLaunchKernelGGL(igso3_grad_kernel, dim3(B_ROWS), dim3(256), 0, stream,
                       sigmas, omegas, pdf_pad, out);
}